// deformable_filter_9689446220041
// MI455X (gfx1250) — compile-verified
//
#include <hip/hip_runtime.h>
#include <hip/hip_bf16.h>
#include <math.h>

typedef __attribute__((ext_vector_type(2))) float v2f;
typedef __attribute__((ext_vector_type(8))) float v8f;

#define Hdim 128
#define Wdim 128
#define HW   16384      // 128*128
#define CIN  64
#define COUT 64
#define NTAP 9
#define KTOT 576        // CIN * NTAP
#define XROW 580        // padded LDS row stride (floats) to avoid bank conflicts

// ---------------------------------------------------------------------------
// Kernel 1: offset = conv3x3(x, w_off) + b_off   -> d_ws as [B][18][H][W]
// ---------------------------------------------------------------------------
__global__ __launch_bounds__(256) void deform_offset_conv_kernel(
    const float* __restrict__ x, const float* __restrict__ w_off,
    const float* __restrict__ b_off, float* __restrict__ offs)
{
    __shared__ float wlds[18 * CIN * NTAP];   // 41472 B
    const int t = threadIdx.x;
    for (int i = t; i < 18 * CIN * NTAP; i += 256) wlds[i] = w_off[i];
    __syncthreads();

    const int pixel = blockIdx.x * 256 + t;
    const int b = pixel >> 14;            // /HW
    const int rem = pixel & (HW - 1);
    const int h = rem >> 7;
    const int w = rem & (Wdim - 1);

    float acc[18];
#pragma unroll
    for (int j = 0; j < 18; ++j) acc[j] = b_off[j];

    const float* xb = x + (size_t)b * CIN * HW;
    for (int c = 0; c < CIN; ++c) {
        const float* xp = xb + c * HW;
        float xv[9];
#pragma unroll
        for (int dx = -1; dx <= 1; ++dx) {
#pragma unroll
            for (int dy = -1; dy <= 1; ++dy) {
                const int hh = h + dx, ww = w + dy;
                const bool ok = ((unsigned)hh < Hdim) && ((unsigned)ww < Wdim);
                xv[(dx + 1) * 3 + (dy + 1)] = ok ? xp[hh * Wdim + ww] : 0.0f;
            }
        }
        const float* wr = &wlds[c * NTAP];     // wlds[(j*64 + c)*9 + n]
#pragma unroll
        for (int j = 0; j < 18; ++j) {
#pragma unroll
            for (int n = 0; n < 9; ++n)
                acc[j] = fmaf(xv[n], wr[j * CIN * NTAP + n], acc[j]);
        }
    }
#pragma unroll
    for (int j = 0; j < 18; ++j)
        offs[(((size_t)b * 18 + j) << 14) + (h << 7) + w] = acc[j];
}

// ---------------------------------------------------------------------------
// Kernel 2: bilinear deformable sampling into LDS + WMMA f32 GEMM
//   workgroup = 16 consecutive pixels (one row segment), 128 thr = 4 waves,
//   wave w handles Cout tile [16w, 16w+16).  D = Wdef(16x576) x Xoff(576x16)
// ---------------------------------------------------------------------------
__global__ __launch_bounds__(128) void deform_gemm_wmma_kernel(
    const float* __restrict__ x, const float* __restrict__ offs,
    const float* __restrict__ w_def, const float* __restrict__ b_def,
    float* __restrict__ out)
{
    __shared__ __align__(16) float s_xoff[16 * XROW];  // 37120 B
    __shared__ float s_wt[144 * 4];                    //  2304 B
    __shared__ int   s_idx[144 * 4];                   //  2304 B

    const int tile = blockIdx.x;        // 4096 tiles
    const int b    = tile >> 10;        // 1024 tiles per batch image
    const int rem  = tile & 1023;
    const int h    = rem >> 3;          // 8 tiles per row (128/16)
    const int w0   = (rem & 7) << 4;
    const int t    = threadIdx.x;

    // ---- Stage 0: bilinear corner indices & weights for 16 pixels x 9 taps
    if (t < 144) {
        const int pix = t / 9, n = t % 9;
        const int w = w0 + pix;
        const float offx = offs[(((size_t)b * 18 + n) << 14) + (h << 7) + w];
        const float offy = offs[(((size_t)b * 18 + n + 9) << 14) + (h << 7) + w];
        const int dx = n / 3 - 1, dy = n % 3 - 1;
        const float px = (float)(h + 1 + dx) + offx;     // padded-frame coords
        const float py = (float)(w + 1 + dy) + offy;
        const float HPm1 = 129.0f;                       // Hp-1 = Wp-1
        const float flx = floorf(px), fly = floorf(py);
        const float qlx = fminf(fmaxf(flx, 0.0f), HPm1);
        const float qly = fminf(fmaxf(fly, 0.0f), HPm1);
        const float qrx = fminf(fmaxf(flx + 1.0f, 0.0f), HPm1);
        const float qry = fminf(fmaxf(fly + 1.0f, 0.0f), HPm1);
        const float pxc = fminf(fmaxf(px, 0.0f), HPm1);
        const float pyc = fminf(fmaxf(py, 0.0f), HPm1);
        const float glt = (1.0f + (qlx - pxc)) * (1.0f + (qly - pyc));
        const float grb = (1.0f - (qrx - pxc)) * (1.0f - (qry - pyc));
        const float glb = (1.0f + (qlx - pxc)) * (1.0f - (qry - pyc));
        const float grt = (1.0f - (qrx - pxc)) * (1.0f + (qly - pyc));
        // padded index -> unpadded; out-of-image corners read x[0] with weight 0
        const int cx[4] = {(int)qlx - 1, (int)qrx - 1, (int)qlx - 1, (int)qrx - 1};
        const int cy[4] = {(int)qly - 1, (int)qry - 1, (int)qry - 1, (int)qly - 1};
        const float cg[4] = {glt, grb, glb, grt};
#pragma unroll
        for (int i = 0; i < 4; ++i) {
            const bool ok = ((unsigned)cx[i] < Hdim) && ((unsigned)cy[i] < Wdim);
            s_idx[t * 4 + i] = ok ? cx[i] * Wdim + cy[i] : 0;
            s_wt[t * 4 + i]  = ok ? cg[i] : 0.0f;
        }
    }
    __syncthreads();

    // ---- Stage 1: gather x_off [16 pixels][K=576] into LDS (shared by 4 waves)
    const float* xb = x + (size_t)b * CIN * HW;
    for (int e = t; e < 16 * KTOT; e += 128) {     // 72 iterations, exact
        const int c   = e / 144;
        const int r2  = e - c * 144;               // pix*9 + n, matches stage-0 slot
        const int pix = r2 / 9;
        const int n   = r2 - pix * 9;
        const float* xp = xb + c * HW;
        const float v = s_wt[r2 * 4 + 0] * xp[s_idx[r2 * 4 + 0]]
                      + s_wt[r2 * 4 + 1] * xp[s_idx[r2 * 4 + 1]]
                      + s_wt[r2 * 4 + 2] * xp[s_idx[r2 * 4 + 2]]
                      + s_wt[r2 * 4 + 3] * xp[s_idx[r2 * 4 + 3]];
        s_xoff[pix * XROW + c * NTAP + n] = v;
    }
    __syncthreads();

    // ---- Stage 2: fp32 WMMA GEMM, one 16-Cout tile per wave
    const int lane   = t & 31;
    const int wave   = t >> 5;          // wave32
    const int o_base = wave << 4;
    const int mn     = lane & 15;       // A row m / B,D column n (pixel)
    const int khalf  = lane >> 4;       // K sub-pair selector per ISA layout
    const float* wrow = w_def + (size_t)(o_base + mn) * KTOT + 2 * khalf;
    const float* brow = s_xoff + mn * XROW + 2 * khalf;

    v8f acc = {};
#pragma unroll 4
    for (int kk = 0; kk < KTOT; kk += 4) {
        const v2f a  = *(const v2f*)(wrow + kk);   // A[m][kk+2*khalf .. +1]
        const v2f bf = *(const v2f*)(brow + kk);   // B[kk+2*khalf .. +1][pixel]
        acc = __builtin_amdgcn_wmma_f32_16x16x4_f32(
            false, a, false, bf, (short)0, acc, false, false);
    }

    // D layout: VGPR r, lane l -> M = r + 8*(l>>4), N = l&15
#pragma unroll
    for (int r = 0; r < 8; ++r) {
        const int o = o_base + r + (khalf << 3);
        out[(((size_t)b * COUT + o) << 14) + (h << 7) + w0 + mn] = acc[r] + b_def[o];
    }
}

// ---------------------------------------------------------------------------
extern "C" void kernel_launch(void* const* d_in, const int* in_sizes, int n_in,
                              void* d_out, int out_size, void* d_ws, size_t ws_size,
                              hipStream_t stream) {
    const float* x     = (const float*)d_in[0];   // [4,64,128,128]
    const float* w_off = (const float*)d_in[1];   // [18,64,3,3]
    const float* b_off = (const float*)d_in[2];   // [18]
    const float* w_def = (const float*)d_in[3];   // [64,64,3,3] -> [64,576]
    const float* b_def = (const float*)d_in[4];   // [64]
    float* out  = (float*)d_out;                  // [4,64,128,128]
    float* offs = (float*)d_ws;                   // [4,18,128,128] scratch

    // Kernel 1: 4*128*128 = 65536 pixels, 256 threads each
    deform_offset_conv_kernel<<<dim3(65536 / 256), dim3(256), 0, stream>>>(
        x, w_off, b_off, offs);

    // Kernel 2: 65536/16 = 4096 pixel tiles, 128 threads (4 waves) each
    deform_gemm_wmma_kernel<<<dim3(4096), dim3(128), 0, stream>>>(
        x, offs, w_def, b_def, out);
}